// A2SI_42296837931755
// MI455X (gfx1250) — compile-verified
//
#include <hip/hip_runtime.h>
#include <math.h>

typedef float v2f __attribute__((ext_vector_type(2)));
typedef float v8f __attribute__((ext_vector_type(8)));

#define LEAKK 0.2f
__device__ __forceinline__ float leakyf(float x){ return x > 0.f ? x : LEAKK * x; }

__device__ __forceinline__ v8f wmma4(v2f a, v2f b, v8f c){
  // V_WMMA_F32_16X16X4_F32 : D = A(16x4,f32) * B(4x16,f32) + C(16x16,f32)
  return __builtin_amdgcn_wmma_f32_16x16x4_f32(false, a, false, b, (short)0, c, false, false);
}

// ---------------- sizes ----------------
#define Bc 4
#define Cc 64
#define Hc 48
#define Wc 48
#define HWc 2304
#define Dc 256
#define Fnc 8
#define Tc 4
#define Vc 16
#define Ec 16
#define Lc 16384

// ---------------- ws layout (float offsets) ----------------
#define OFF_W3BUF  0u          // 32*36864 = 1179648
#define OFF_W1BUF  1179648u    // 32*4096  = 131072
#define OFF_Y1     1310720u    // 4*64*2304= 589824
#define OFF_BVF    1900544u    // 32*256   = 8192
#define OFF_FFCAT  1908736u    // 4*256    = 1024
#define OFF_FFHID  1909760u    // 4*64     = 256
#define OFF_FFOUT  1910016u    // 4*256    = 1024
#define OFF_ATHID  1911040u    // 4*64     = 256
#define OFF_ATOUT  1911296u    // 4*128    = 512
#define OFF_F1V    1911808u    // 4*16     = 64
#define OFF_IESUM  1911872u    // 4*128    = 512
#define OFF_RMS    1912384u    // 4
#define OFF_M1HID  1912448u    // 32*64    = 2048
#define OFF_M3HID  1914496u    // 32*64    = 2048
#define OFF_IDX    1916544u    // 32 ints

// ================= per-batch RMS of task_f =================
__global__ void k_rms(const float* __restrict__ task_f, float* __restrict__ rms){
  int b = blockIdx.x;
  const float* x = task_f + b * Lc;
  float s = 0.f;
  for (int i = threadIdx.x; i < Lc; i += 256){ float v = x[i]; s += v * v; }
  __shared__ float red[256];
  red[threadIdx.x] = s; __syncthreads();
  for (int k = 128; k > 0; k >>= 1){ if (threadIdx.x < k) red[threadIdx.x] += red[threadIdx.x + k]; __syncthreads(); }
  if (threadIdx.x == 0) rms[b] = sqrtf(red[0] / (float)Lc);
}

// ===== conv1d(leaky(conv1d(x))) -> mean(L)  via channel-sum trick =====
template<int CI, int CO>
__global__ void k_tconv(const float* __restrict__ xall, const float* __restrict__ w1,
                        const float* __restrict__ w2, const float* __restrict__ rms,
                        float* __restrict__ out, int ostride, int ooff){
  int b = blockIdx.x;
  const float* x = xall + b * Lc;
  float inv = 1.f;
  if (rms) inv = 1.f / (rms[b] + 1e-8f);
  float s[CI];
  #pragma unroll
  for (int i = 0; i < CI; i++) s[i] = 0.f;
  __shared__ float S[CI], A0[CI], AL[CI];
  if (threadIdx.x < CI) S[threadIdx.x] = 0.f;
  __syncthreads();
  for (int l = threadIdx.x; l < Lc; l += blockDim.x){
    float xm = (l > 0)      ? x[l-1] * inv : 0.f;
    float x0 = x[l] * inv;
    float xp = (l < Lc - 1) ? x[l+1] * inv : 0.f;
    bool first = (l == 0), last = (l == Lc - 1);
    #pragma unroll 4
    for (int i = 0; i < CI; i++){
      float a = leakyf(w1[i*3+0]*xm + w1[i*3+1]*x0 + w1[i*3+2]*xp);
      s[i] += a;
      if (first) A0[i] = a;
      if (last)  AL[i] = a;
    }
  }
  for (int i = 0; i < CI; i++) atomicAdd(&S[i], s[i]);
  __syncthreads();
  for (int o = threadIdx.x; o < CO; o += blockDim.x){
    float acc = 0.f;
    for (int i = 0; i < CI; i++){
      float Si = S[i];
      const float* w = w2 + (o*CI + i)*3;
      acc += w[0]*(Si - AL[i]) + w[1]*Si + w[2]*(Si - A0[i]);
    }
    out[b*ostride + ooff + o] = acc * (1.f / (float)Lc);
  }
}

// ======= iA branch: mean_{hw} leaky(bn(conv3x3(img))) -> i1 (B,8) =======
__global__ void k_iA(const float* __restrict__ img, const float* __restrict__ w,
                     const float* __restrict__ bn, float* __restrict__ out,
                     int ostride, int ooff){
  int o = blockIdx.x, b = blockIdx.y;  // grid (8, B)
  float g = bn[0*8+o], bb = bn[1*8+o], mm = bn[2*8+o], vv = bn[3*8+o];
  float scale = g * rsqrtf(vv + 1e-5f);
  const float* xb = img + b * Cc * HWc;
  const float* wo = w + o * Cc * 9;
  float part = 0.f;
  for (int p = threadIdx.x; p < HWc; p += 256){
    int h = p / Wc, wq = p % Wc;
    float acc = 0.f;
    for (int c = 0; c < Cc; c++){
      const float* xc = xb + c * HWc;
      const float* wc = wo + c * 9;
      #pragma unroll
      for (int kh = 0; kh < 3; kh++){
        int hh = h + kh - 1; if (hh < 0 || hh >= Hc) continue;
        #pragma unroll
        for (int kw = 0; kw < 3; kw++){
          int ww = wq + kw - 1; if (ww < 0 || ww >= Wc) continue;
          acc += wc[kh*3+kw] * xc[hh*Wc + ww];
        }
      }
    }
    part += leakyf((acc - mm) * scale + bb);
  }
  __shared__ float red[256];
  red[threadIdx.x] = part; __syncthreads();
  for (int k = 128; k > 0; k >>= 1){ if (threadIdx.x < k) red[threadIdx.x] += red[threadIdx.x + k]; __syncthreads(); }
  if (threadIdx.x == 0) out[b*ostride + ooff + o] = red[0] * (1.f / (float)HWc);
}

// ======= generic: out(row,64) = relu(LN(act(row,K) @ W(K,64) + b)) =======
__global__ void k_lin_ln_relu(const float* __restrict__ act, int K,
                              const float* __restrict__ Wm, const float* __restrict__ bias,
                              const float* __restrict__ g, const float* __restrict__ beta,
                              float* __restrict__ out){
  int row = blockIdx.x, j = threadIdx.x;  // 64 threads
  const float* a = act + row * K;
  float acc = bias[j];
  for (int k = 0; k < K; k++) acc += a[k] * Wm[k*64 + j];
  __shared__ float r1[64];
  __shared__ float mu, var_;
  r1[j] = acc; __syncthreads();
  for (int k = 32; k > 0; k >>= 1){ if (j < k) r1[j] += r1[j + k]; __syncthreads(); }
  if (j == 0) mu = r1[0] / 64.f;
  __syncthreads();
  float d = acc - mu;
  r1[j] = d * d; __syncthreads();
  for (int k = 32; k > 0; k >>= 1){ if (j < k) r1[j] += r1[j + k]; __syncthreads(); }
  if (j == 0) var_ = r1[0] / 64.f;
  __syncthreads();
  float y = (acc - mu) * rsqrtf(var_ + 1e-5f) * g[j] + beta[j];
  out[row*64 + j] = fmaxf(y, 0.f);
}

// ======= WMMA GEMM: out(M,N) = act(M,64) @ W(64,N) + bias(N)  (K fixed 64) =======
// FULL: M is a multiple of 16 -> no A-mask in the hot loop, no store guards.
template<bool FULL>
__global__ void k_gemm64(const float* __restrict__ act, int M,
                         const float* __restrict__ Wm, int N,
                         const float* __restrict__ bias, float* __restrict__ out){
  // grid: (N/16, ceil(M/16)), block: 32 (one wave, EXEC all ones)
  int nt = blockIdx.x * 16, mt = blockIdx.y * 16;
  int lane = threadIdx.x & 31;
  int half = lane >> 4, lid = lane & 15;
  int mA = mt + lid;
  int nB = nt + lid;
  const float* arow;
  float msk = 1.f;
  if (FULL){
    arow = act + mA * 64;
  } else {
    int mclamp = (mA < M) ? mA : (M - 1);
    msk  = (mA < M) ? 1.f : 0.f;
    arow = act + mclamp * 64;
  }
  const float* wcol = Wm + nB;
  v8f acc = {};
  #pragma unroll 4
  for (int k0 = 0; k0 < 64; k0 += 4){
    int kk = k0 + half * 2;
    v2f a, bv;
    if (FULL){
      a.x = arow[kk];
      a.y = arow[kk + 1];
    } else {
      a.x = arow[kk]     * msk;
      a.y = arow[kk + 1] * msk;
    }
    bv.x = wcol[kk*N];
    bv.y = wcol[(kk+1)*N];
    acc = wmma4(a, bv, acc);
  }
  float bi = bias[nB];           // hoisted: load once, not per store guard
  #pragma unroll
  for (int r = 0; r < 8; r++){
    int m = mt + r + half * 8;
    if (FULL || m < M) out[m*N + nB] = acc[r] + bi;
  }
}

// ======= Threefry-2x32 (JAX-compatible) + gumbel =======
__device__ __forceinline__ void threefry2x32(unsigned k0, unsigned k1, unsigned x0, unsigned x1,
                                             unsigned* o0, unsigned* o1){
  const unsigned R0[4] = {13,15,26,6};
  const unsigned R1[4] = {17,29,16,24};
  unsigned ks[3]; ks[0] = k0; ks[1] = k1; ks[2] = k0 ^ k1 ^ 0x1BD11BDAu;
  x0 += ks[0]; x1 += ks[1];
  #pragma unroll
  for (int g = 0; g < 5; g++){
    const unsigned* R = (g & 1) ? R1 : R0;
    #pragma unroll
    for (int r = 0; r < 4; r++){
      x0 += x1;
      x1 = (x1 << R[r]) | (x1 >> (32 - R[r]));
      x1 ^= x0;
    }
    x0 += ks[(g + 1) % 3];
    x1 += ks[(g + 2) % 3] + (unsigned)(g + 1);
  }
  *o0 = x0; *o1 = x1;
}

__device__ __forceinline__ float gumbel_at(int n){
  unsigned c0 = (n < 64) ? (unsigned)n : (unsigned)(n - 64);
  unsigned c1 = c0 + 64u;
  unsigned o0, o1;
  threefry2x32(0u, 42u, c0, c1, &o0, &o1);
  unsigned bits = (n < 64) ? o0 : o1;
  float f = __uint_as_float((bits >> 9) | 0x3f800000u) - 1.0f;
  float u = f * (1.0f - 1e-10f) + 1e-10f;
  u = fmaxf(u, 1e-10f);
  return -logf(-logf(u));
}

// ======= attention head: logits + gumbel -> idx (B,Fn) =======
__global__ void k_att(const float* __restrict__ atout /*(4,128)*/,
                      const float* __restrict__ protos /*(4,16)*/,
                      int* __restrict__ idxout){
  __shared__ float pn[Tc][Ec];
  int tid = threadIdx.x;  // 64 threads
  if (tid < Tc){
    float ss = 0.f;
    for (int e = 0; e < Ec; e++){ float v = protos[tid*Ec + e]; ss += v * v; }
    float inv = 1.f / fmaxf(sqrtf(ss), 1e-8f);
    for (int e = 0; e < Ec; e++) pn[tid][e] = protos[tid*Ec + e] * inv;
  }
  __syncthreads();
  if (tid < Bc * Fnc){
    int b = tid / Fnc, f = tid % Fnc;
    float col[Ec]; float ss = 0.f;
    for (int e = 0; e < Ec; e++){ float v = atout[b*128 + e*Fnc + f]; col[e] = v; ss += v * v; }
    float inv = 1.f / fmaxf(sqrtf(ss), 1e-8f);
    int best = 0; float bestv = -1e30f;
    for (int t = 0; t < Tc; t++){
      float lg = 0.f;
      for (int e = 0; e < Ec; e++) lg += pn[t][e] * col[e] * inv;
      float z = lg + gumbel_at((b*Fnc + f)*Tc + t);  // softmax monotone -> argmax unchanged
      if (z > bestv){ bestv = z; best = t; }
    }
    idxout[b*Fnc + f] = best;
  }
}

__global__ void k_finie(const float* __restrict__ iesum, float* __restrict__ ffcat){
  int i = threadIdx.x + blockIdx.x * blockDim.x;
  if (i < Bc * 128){
    int b = i / 128, o = i % 128;
    ffcat[b*256 + 128 + o] = iesum[i] * (1.f / (float)HWc);
  }
}

// ======= WMMA 3x3 conv, pad 1 (im2col from LDS halo tile, tap-major K order) =======
// STORE: write leaky(bn(y)); else accumulate mean sums with atomics (for ie)
template<int CO, bool STORE>
__global__ void k_conv3(const float* __restrict__ in, const float* __restrict__ wgt,
                        const float* __restrict__ bn, float* __restrict__ outp){
  __shared__ float xs[Cc * 3 * 18];  // halo tile: 64ch x rows[h-1..h+1] x cols[w0-1..w0+16]
  int blk = blockIdx.x;
  int b = blk / 144, tile = blk % 144;
  int h = tile / 3, w0 = (tile % 3) * 16;
  const float* xb = in + b * Cc * HWc;
  int tcount = CO * 2;
  for (int i = threadIdx.x; i < Cc * 3 * 18; i += tcount){
    int c = i / 54, rem = i % 54, r = rem / 18, col = rem % 18;
    int hh = h + r - 1, ww = w0 + col - 1;
    xs[i] = (hh >= 0 && hh < Hc && ww >= 0 && ww < Wc) ? xb[c*HWc + hh*Wc + ww] : 0.f;
  }
  __syncthreads();
  int wave = threadIdx.x >> 5, lane = threadIdx.x & 31;
  int half = lane >> 4, lid = lane & 15;
  int mbase = wave * 16;
  const float* wrow = wgt + (mbase + lid) * (Cc * 9);
  __builtin_prefetch(wrow, 0, 0);
  v8f acc = {};
  // K reordered tap-major: k' = j*64 + c  (reduction order is free)
  #pragma unroll
  for (int j = 0; j < 9; j++){
    const int off = (j/3)*18 + (j%3) + lid;  // constant-folded per unrolled j
    const float* wj = wrow + j;              // stride 9 over c
    #pragma unroll 4
    for (int k0 = 0; k0 < Cc; k0 += 4){
      int kk = k0 + half * 2;
      v2f a, bv;
      a.x = wj[kk*9];
      a.y = wj[(kk+1)*9];
      bv.x = xs[kk*54 + off];
      bv.y = xs[(kk+1)*54 + off];
      acc = wmma4(a, bv, acc);
    }
  }
  #pragma unroll
  for (int r = 0; r < 8; r++){
    int ch = mbase + r + half * 8;
    float g = bn[0*CO+ch], bb = bn[1*CO+ch], mm = bn[2*CO+ch], vv = bn[3*CO+ch];
    float y = leakyf((acc[r] - mm) * (g * rsqrtf(vv + 1e-5f)) + bb);
    if (STORE){
      outp[b*CO*HWc + ch*HWc + h*Wc + w0 + lid] = y;
    } else {
      acc[r] = y;
    }
  }
  if (!STORE){
    #pragma unroll
    for (int r = 0; r < 8; r++){
      float s = acc[r];
      for (int m = 1; m < 16; m <<= 1) s += __shfl_xor(s, m, 32);  // reduce within 16-lane halves
      if (lid == 0){
        int ch = mbase + r + half * 8;
        atomicAdd(&outp[b*CO + ch], s);
      }
    }
  }
}

// ======= fused dynamic convs + blr bn/leaky + mean_f + zc 1x1 + residual =======
__global__ void k_dp(const float* __restrict__ img, const float* __restrict__ w1buf,
                     const float* __restrict__ w3buf, const int* __restrict__ idx,
                     const float* __restrict__ blr, const float* __restrict__ zcw,
                     const float* __restrict__ zcb, float* __restrict__ outp){
  __shared__ float xs[Cc * 7 * 22];   // halo for dilation<=3: rows[h-3..h+3] cols[w0-3..w0+18]
  __shared__ float xsh[Cc * 16];      // mean-over-f tile for zc epilogue
  int blk = blockIdx.x;
  int b = blk / 144, tile = blk % 144;
  int h = tile / 3, w0 = (tile % 3) * 16;
  const float* xb = img + b * Cc * HWc;
  for (int i = threadIdx.x; i < Cc * 7 * 22; i += 128){
    int c = i / 154, rem = i % 154, r = rem / 22, col = rem % 22;
    int hh = h + r - 3, ww = w0 + col - 3;
    xs[i] = (hh >= 0 && hh < Hc && ww >= 0 && ww < Wc) ? xb[c*HWc + hh*Wc + ww] : 0.f;
  }
  __syncthreads();
  int wave = threadIdx.x >> 5, lane = threadIdx.x & 31;
  int half = lane >> 4, lid = lane & 15;
  int mbase = wave * 16;
  float accum[8];
  #pragma unroll
  for (int r = 0; r < 8; r++) accum[r] = 0.f;
  for (int f = 0; f < Fnc; f++){
    int t = idx[b*Fnc + f];  // uniform across block
    // prefetch next dynamic-weight row while this one computes
    if (f + 1 < Fnc){
      int tn = idx[b*Fnc + f + 1];
      const float* np = (tn == 0) ? (w1buf + (b*Fnc + f + 1)*4096 + (mbase + lid)*64)
                                  : (w3buf + (b*Fnc + f + 1)*36864 + (mbase + lid)*576);
      __builtin_prefetch(np, 0, 0);
    }
    v8f acc = {};
    if (t == 0){
      const float* Wp = w1buf + (b*Fnc + f)*4096 + (mbase + lid)*64;
      #pragma unroll 4
      for (int k0 = 0; k0 < 64; k0 += 4){
        int kk = k0 + half * 2;
        v2f a, bv;
        a.x = Wp[kk]; a.y = Wp[kk + 1];
        bv.x = xs[kk*154     + 3*22 + 3 + lid];
        bv.y = xs[(kk+1)*154 + 3*22 + 3 + lid];
        acc = wmma4(a, bv, acc);
      }
    } else {
      int d = t;  // dilation 1..3, pad=d
      const float* Wp = w3buf + (b*Fnc + f)*36864 + (mbase + lid)*576;
      // tap-major K order: no div/mod in the hot loop
      #pragma unroll
      for (int j = 0; j < 9; j++){
        const int off = (3 + (j/3 - 1)*d)*22 + 3 + lid + (j%3 - 1)*d;
        const float* wj = Wp + j;
        #pragma unroll 4
        for (int k0 = 0; k0 < Cc; k0 += 4){
          int kk = k0 + half * 2;
          v2f a, bv;
          a.x = wj[kk*9];
          a.y = wj[(kk+1)*9];
          bv.x = xs[kk*154 + off];
          bv.y = xs[(kk+1)*154 + off];
          acc = wmma4(a, bv, acc);
        }
      }
    }
    #pragma unroll
    for (int r = 0; r < 8; r++){
      int ch = mbase + r + half * 8;
      float g = blr[0*Cc+ch], bb = blr[1*Cc+ch], mm = blr[2*Cc+ch], vv = blr[3*Cc+ch];
      accum[r] += leakyf((acc[r] - mm) * (g * rsqrtf(vv + 1e-5f)) + bb);
    }
  }
  #pragma unroll
  for (int r = 0; r < 8; r++){
    int ch = mbase + r + half * 8;
    xsh[ch*16 + lid] = accum[r] * (1.f / (float)Fnc);
  }
  __syncthreads();
  v8f acc2 = {};
  #pragma unroll 4
  for (int k0 = 0; k0 < 64; k0 += 4){
    int kk = k0 + half * 2;
    v2f a, bv;
    a.x = zcw[(mbase + lid)*64 + kk];
    a.y = zcw[(mbase + lid)*64 + kk + 1];
    bv.x = xsh[kk*16 + lid];
    bv.y = xsh[(kk+1)*16 + lid];
    acc2 = wmma4(a, bv, acc2);
  }
  #pragma unroll
  for (int r = 0; r < 8; r++){
    int ch = mbase + r + half * 8;
    int p = h*Wc + w0 + lid;
    outp[b*Cc*HWc + ch*HWc + p] = acc2[r] + zcb[ch] + img[b*Cc*HWc + ch*HWc + p];
  }
}

// ======= selection logic: Bg / scores / argsort / rank -> bvf (32,256) =======
__global__ void k_select(const float* __restrict__ bank, const int* __restrict__ idx,
                         const float* __restrict__ ff, float* __restrict__ bvf){
  int b = blockIdx.x, tid = threadIdx.x;  // 256 threads
  __shared__ float ffn[Dc];
  __shared__ float red[256];
  __shared__ float sc[Tc][Vc];
  __shared__ int order[Tc][Vc];
  __shared__ int tb[Tc];
  __shared__ int selbuf[Fnc];
  float v = ff[b*Dc + tid];
  red[tid] = v * v; __syncthreads();
  for (int k = 128; k > 0; k >>= 1){ if (tid < k) red[tid] += red[tid + k]; __syncthreads(); }
  float inv = 1.f / fmaxf(sqrtf(red[0]), 1e-12f);
  ffn[tid] = v * inv;
  if (tid < Tc){
    int t = tid, first = 0;
    for (int f = Fnc - 1; f >= 0; f--) if (idx[b*Fnc + f] == t) first = f;
    tb[t] = idx[b*Fnc + first];  // == t if present, else idx[b,0]
  }
  __syncthreads();
  if (tid < Tc * Vc){
    int t = tid / Vc, vv = tid % Vc;
    const float* row = bank + tb[t]*(Vc*Dc) + vv*Dc;
    float dot = 0.f, ss = 0.f;
    for (int d = 0; d < Dc; d++){ float r = row[d]; dot += ffn[d] * r; ss += r * r; }
    sc[t][vv] = dot / fmaxf(sqrtf(ss), 1e-12f);
  }
  __syncthreads();
  if (tid < Tc){
    int t = tid;
    bool used[Vc]; float s[Vc];
    for (int i = 0; i < Vc; i++){ used[i] = false; s[i] = sc[t][i]; }
    for (int r = 0; r < Vc; r++){       // stable descending argsort (ties -> lower index)
      int bi = 0; float bvl = -1e30f; bool found = false;
      for (int i = 0; i < Vc; i++) if (!used[i] && (!found || s[i] > bvl)){ bvl = s[i]; bi = i; found = true; }
      used[bi] = true; order[t][r] = bi;
    }
  }
  __syncthreads();
  if (tid < Fnc){
    int f = tid, t = idx[b*Fnc + f];
    int rank = 0;
    for (int fp = 0; fp < f; fp++) if (idx[b*Fnc + fp] == t) rank++;
    selbuf[f] = order[t][rank];
  }
  __syncthreads();
  for (int f = 0; f < Fnc; f++){
    int t = idx[b*Fnc + f];
    const float* src = bank + tb[t]*(Vc*Dc) + selbuf[f]*Dc;
    bvf[(b*Fnc + f)*Dc + tid] = src[tid];
  }
}

// =====================================================================
extern "C" void kernel_launch(void* const* d_in, const int* in_sizes, int n_in,
                              void* d_out, int out_size, void* d_ws, size_t ws_size,
                              hipStream_t stream){
  (void)in_sizes; (void)n_in; (void)out_size; (void)ws_size;
  const float* bank   = (const float*)d_in[0];
  const float* taskf  = (const float*)d_in[1];
  const float* img    = (const float*)d_in[2];
  const float* protos = (const float*)d_in[3];
  const float* tA_w1  = (const float*)d_in[4];
  const float* tA_w2  = (const float*)d_in[5];
  const float* iA_w   = (const float*)d_in[6];
  const float* iA_bn  = (const float*)d_in[7];
  const float* at_w1  = (const float*)d_in[8];
  const float* at_b1  = (const float*)d_in[9];
  const float* at_g   = (const float*)d_in[10];
  const float* at_be  = (const float*)d_in[11];
  const float* at_w2  = (const float*)d_in[12];
  const float* at_b2  = (const float*)d_in[13];
  const float* bv_t1  = (const float*)d_in[14];
  const float* bv_t2  = (const float*)d_in[15];
  const float* bv_ic1 = (const float*)d_in[16];
  const float* bv_bn1 = (const float*)d_in[17];
  const float* bv_ic2 = (const float*)d_in[18];
  const float* bv_bn2 = (const float*)d_in[19];
  const float* bv_w1  = (const float*)d_in[20];
  const float* bv_b1  = (const float*)d_in[21];
  const float* bv_g   = (const float*)d_in[22];
  const float* bv_be  = (const float*)d_in[23];
  const float* bv_w2  = (const float*)d_in[24];
  const float* bv_b2  = (const float*)d_in[25];
  const float* m1_w1  = (const float*)d_in[26];
  const float* m1_b1  = (const float*)d_in[27];
  const float* m1_g   = (const float*)d_in[28];
  const float* m1_be  = (const float*)d_in[29];
  const float* m1_w2  = (const float*)d_in[30];
  const float* m1_b2  = (const float*)d_in[31];
  const float* m3_w1  = (const float*)d_in[32];
  const float* m3_b1  = (const float*)d_in[33];
  const float* m3_g   = (const float*)d_in[34];
  const float* m3_be  = (const float*)d_in[35];
  const float* m3_w2  = (const float*)d_in[36];
  const float* m3_b2  = (const float*)d_in[37];
  const float* blr_bn = (const float*)d_in[38];
  const float* zc_w   = (const float*)d_in[39];
  const float* zc_b   = (const float*)d_in[40];

  float* ws    = (float*)d_ws;
  float* w3buf = ws + OFF_W3BUF;
  float* w1buf = ws + OFF_W1BUF;
  float* y1    = ws + OFF_Y1;
  float* bvf   = ws + OFF_BVF;
  float* ffcat = ws + OFF_FFCAT;
  float* ffhid = ws + OFF_FFHID;
  float* ffout = ws + OFF_FFOUT;
  float* athid = ws + OFF_ATHID;
  float* atout = ws + OFF_ATOUT;
  float* f1v   = ws + OFF_F1V;
  float* iesum = ws + OFF_IESUM;
  float* rmsv  = ws + OFF_RMS;
  float* m1hid = ws + OFF_M1HID;
  float* m3hid = ws + OFF_M3HID;
  int*   idxb  = (int*)(ws + OFF_IDX);

  hipMemsetAsync(iesum, 0, Bc * 128 * sizeof(float), stream);

  // ---- scalar prologue ----
  k_rms<<<Bc, 256, 0, stream>>>(taskf, rmsv);
  k_tconv<4, 8><<<Bc, 256, 0, stream>>>(taskf, tA_w1, tA_w2, nullptr, f1v, 16, 0);      // t1
  k_tconv<64, 128><<<Bc, 256, 0, stream>>>(taskf, bv_t1, bv_t2, rmsv, ffcat, 256, 0);   // te
  k_iA<<<dim3(8, Bc), 256, 0, stream>>>(img, iA_w, iA_bn, f1v, 16, 8);                  // i1

  // ---- attention head -> idx ----
  k_lin_ln_relu<<<Bc, 64, 0, stream>>>(f1v, 16, at_w1, at_b1, at_g, at_be, athid);
  k_gemm64<false><<<dim3(128/16, 1), 32, 0, stream>>>(athid, Bc, at_w2, 128, at_b2, atout);
  k_att<<<1, 64, 0, stream>>>(atout, protos, idxb);

  // ---- bv image branch (big WMMA convs) ----
  k_conv3<64, true ><<<Bc * 144, 128, 0, stream>>>(img, bv_ic1, bv_bn1, y1);
  k_conv3<128, false><<<Bc * 144, 256, 0, stream>>>(y1, bv_ic2, bv_bn2, iesum);
  k_finie<<<2, 256, 0, stream>>>(iesum, ffcat);

  // ---- ff MLP ----
  k_lin_ln_relu<<<Bc, 64, 0, stream>>>(ffcat, 256, bv_w1, bv_b1, bv_g, bv_be, ffhid);
  k_gemm64<false><<<dim3(256/16, 1), 32, 0, stream>>>(ffhid, Bc, bv_w2, 256, bv_b2, ffout);

  // ---- selection -> bvf ----
  k_select<<<Bc, 256, 0, stream>>>(bank, idxb, ffout, bvf);

  // ---- weight generators (WMMA GEMMs, M=32 -> FULL path) ----
  k_lin_ln_relu<<<32, 64, 0, stream>>>(bvf, 256, m1_w1, m1_b1, m1_g, m1_be, m1hid);
  k_gemm64<true><<<dim3(4096/16, 2), 32, 0, stream>>>(m1hid, 32, m1_w2, 4096, m1_b2, w1buf);
  k_lin_ln_relu<<<32, 64, 0, stream>>>(bvf, 256, m3_w1, m3_b1, m3_g, m3_be, m3hid);
  k_gemm64<true><<<dim3(36864/16, 2), 32, 0, stream>>>(m3hid, 32, m3_w2, 36864, m3_b2, w3buf);

  // ---- fused dynamic conv + bn/leaky + mean_f + zc 1x1 + residual ----
  k_dp<<<Bc * 144, 128, 0, stream>>>(img, w1buf, w3buf, idxb, blr_bn, zc_w, zc_b, (float*)d_out);
}